// FUB_18193481466135
// MI455X (gfx1250) — compile-verified
//
#include <hip/hip_runtime.h>

typedef __attribute__((ext_vector_type(2))) float v2f;
typedef __attribute__((ext_vector_type(8))) float v8f;

#define N_IMG 64
#define C_CH 256
#define FMAP 4096   // 256 * 16
#define PIX 16

// ---------------- Kernel 1: agg[j] = sum_n x[n][j] / 64 ----------------
__global__ void agg_kernel(const float* __restrict__ x, float* __restrict__ agg) {
    int j = blockIdx.x * blockDim.x + threadIdx.x;  // 0..4095
    float s = 0.f;
    #pragma unroll 8
    for (int n = 0; n < N_IMG; ++n) s += x[n * FMAP + j];
    agg[j] = s * (1.0f / 64.0f);
}

// ------- Kernel 2: split-K GEMV partials[chunk][j] = sum_{k in chunk} agg[k]*Wg[k][j] -------
// grid: 256 blocks = 64 K-chunks (64 k each) x 4 j-blocks (1024 j each, float4 per thread)
__global__ void gemv_partial(const float* __restrict__ Wg, const float* __restrict__ agg,
                             float* __restrict__ partials) {
    int chunk = blockIdx.x >> 2;          // 0..63
    int jb    = blockIdx.x & 3;           // 0..3
    int j     = jb * 1024 + threadIdx.x * 4;
    int k0    = chunk * 64;
    float4 acc = make_float4(0.f, 0.f, 0.f, 0.f);
    for (int k = k0; k < k0 + 64; ++k) {
        float a = agg[k];
        float4 w = *(const float4*)(Wg + (size_t)k * FMAP + j);
        acc.x = fmaf(a, w.x, acc.x);
        acc.y = fmaf(a, w.y, acc.y);
        acc.z = fmaf(a, w.z, acc.z);
        acc.w = fmaf(a, w.w, acc.w);
    }
    *(float4*)(partials + chunk * FMAP + j) = acc;
}

// ------- Kernel 3: rst[j] = b_g[j] + sum_chunk partials; out[n][j] = rst[j] -------
__global__ void finalize_out(const float* __restrict__ partials, const float* __restrict__ bg,
                             float* __restrict__ out) {
    int j = blockIdx.x * blockDim.x + threadIdx.x;  // 0..4095
    float s = bg[j];
    #pragma unroll 8
    for (int c = 0; c < 64; ++c) s += partials[c * FMAP + j];
    for (int n = 0; n < N_IMG; ++n) out[n * FMAP + j] = s;
}

// ------- Kernel 4: per-image channel-mixing GEMMs via V_WMMA_F32_16X16X4_F32 -------
// One wave per (img, oTile): U = (W1+W2) x[img], V = W2 x[img]  (per-pixel 256x256 matmul)
// P[img][o][p] = x - U - b_e ; Q[img][o][p] = V
__global__ void pq_wmma_kernel(const float* __restrict__ x, const float* __restrict__ We,
                               const float* __restrict__ be,
                               float* __restrict__ P, float* __restrict__ Q) {
    const int lane   = threadIdx.x & 31;
    const int waveId = blockIdx.x * (blockDim.x >> 5) + (threadIdx.x >> 5);  // 0..1023
    const int img    = waveId >> 4;          // 0..63
    const int o0     = (waveId & 15) << 4;   // 0,16,...,240
    const int m      = lane & 15;            // row (A) / column (B,C/D)
    const int hiHalf = lane >> 4;            // lanes 16-31 handle K+2 / rows M+8
    const int k0     = hiHalf * 2;           // K base inside each group of 4

    v8f accU = {};
    v8f accV = {};
    const float* xImg  = x + img * FMAP;
    const float* weRow = We + (o0 + m) * (2 * C_CH);

    for (int c = 0; c < C_CH; c += 4) {
        // A operand (16x4 f32): VGPR0 = A[m][k0], VGPR1 = A[m][k0+1]
        float w1a = weRow[c + k0];
        float w1b = weRow[c + k0 + 1];
        float w2a = weRow[C_CH + c + k0];
        float w2b = weRow[C_CH + c + k0 + 1];
        v2f aU; aU[0] = w1a + w2a; aU[1] = w1b + w2b;   // W1 + W2
        v2f aV; aV[0] = w2a;       aV[1] = w2b;          // W2
        // B operand (4x16 f32): B[k][p] = x[img][c+k][p]; column p = m
        v2f b;
        b[0] = xImg[(c + k0) * PIX + m];
        b[1] = xImg[(c + k0 + 1) * PIX + m];
        // D = A*B + C  (8 args: neg_a, A, neg_b, B, c_mod, C, reuse_a, reuse_b)
        accU = __builtin_amdgcn_wmma_f32_16x16x4_f32(false, aU, false, b,
                                                     (short)0, accU, false, false);
        accV = __builtin_amdgcn_wmma_f32_16x16x4_f32(false, aV, false, b,
                                                     (short)0, accV, false, false);
    }

    // C/D layout: VGPR r -> row o0 + r + 8*hiHalf, column m
    #pragma unroll
    for (int r = 0; r < 8; ++r) {
        int o   = o0 + r + 8 * hiHalf;
        int idx = img * FMAP + o * PIX + m;
        P[idx] = xImg[o * PIX + m] - accU[r] - be[o];
        Q[idx] = accV[r];
    }
}

// ------- Kernel 5: edges[i*64+j] = (i==j) ? 1 : sum_k |P[j][k] - Q[i][k]| -------
__global__ void edges_kernel(const float* __restrict__ P, const float* __restrict__ Q,
                             float* __restrict__ edges) {
    const int lane   = threadIdx.x & 31;
    const int waveId = blockIdx.x * (blockDim.x >> 5) + (threadIdx.x >> 5);  // 0..4095
    const int i = waveId >> 6;
    const int j = waveId & 63;
    const float4* p4 = (const float4*)(P + j * FMAP);
    const float4* q4 = (const float4*)(Q + i * FMAP);
    float s = 0.f;
    for (int k = lane; k < FMAP / 4; k += 32) {
        float4 a = p4[k];
        float4 b = q4[k];
        s += fabsf(a.x - b.x) + fabsf(a.y - b.y) + fabsf(a.z - b.z) + fabsf(a.w - b.w);
    }
    #pragma unroll
    for (int off = 16; off >= 1; off >>= 1) s += __shfl_xor(s, off, 32);
    if (lane == 0) edges[i * 64 + j] = (i == j) ? 1.0f : s;
}

extern "C" void kernel_launch(void* const* d_in, const int* in_sizes, int n_in,
                              void* d_out, int out_size, void* d_ws, size_t ws_size,
                              hipStream_t stream) {
    const float* x  = (const float*)d_in[0];  // (64,1,256,4,4)
    const float* We = (const float*)d_in[1];  // (256,512)
    const float* be = (const float*)d_in[2];  // (256,)
    const float* Wg = (const float*)d_in[3];  // (4096,4096)
    const float* bg = (const float*)d_in[4];  // (4096,)

    float* out   = (float*)d_out;             // 64*4096 floats
    float* edges = (float*)d_out + N_IMG * FMAP;  // 4096 floats

    float* ws       = (float*)d_ws;
    float* agg      = ws;                         // 4096 floats
    float* partials = agg + FMAP;                 // 64*4096 floats
    float* P        = partials + 64 * FMAP;       // 64*4096 floats
    float* Q        = P + N_IMG * FMAP;           // 64*4096 floats

    // ---- Graph path 1: out = broadcast(agg @ W_g + b_g) ----
    agg_kernel<<<FMAP / 256, 256, 0, stream>>>(x, agg);
    gemv_partial<<<256, 256, 0, stream>>>(Wg, agg, partials);
    finalize_out<<<FMAP / 256, 256, 0, stream>>>(partials, bg, out);

    // ---- Graph path 2: WMMA channel-mixing -> L1 distance matrix ----
    pq_wmma_kernel<<<128, 256, 0, stream>>>(x, We, be, P, Q);   // 1024 waves
    edges_kernel<<<512, 256, 0, stream>>>(P, Q, edges);          // 4096 waves
}